// TangentImagesToEquirectangular_58463094833212
// MI455X (gfx1250) — compile-verified
//
#include <hip/hip_runtime.h>
#include <hip/hip_bf16.h>

// Problem constants from the reference
#define B_ 4
#define C_ 3
#define F_ 80
#define N_ 256
#define H_ 1024
#define W_ 2048

#define BC_  (B_ * C_)          // 12 channel planes
#define FNN_ (F_ * N_ * N_)     // 5,242,880 floats per (b,c) plane
#define HW_  (H_ * W_)          // 2,097,152 pixels

// Native clang vector type (accepted by __builtin_nontemporal_load, unlike
// HIP's float2 class type).
typedef float v2f __attribute__((ext_vector_type(2)));

// One thread per output pixel. Index/weight math done once, reused across all
// 12 (b,c) channel planes. 48 independent gather loads are issued per thread
// (full unroll) to maximize memory-level parallelism against HBM latency.
// Streaming traffic (quad, uv, out) uses non-temporal hints so the 192 MB L2
// stays dedicated to the 252 MB gather atlas.
__global__ __launch_bounds__(256) void
tangent_to_equirect_bilinear(const float* __restrict__ x,
                             const int*   __restrict__ quad,
                             const float* __restrict__ uv,
                             float*       __restrict__ out) {
  const int p = blockIdx.x * blockDim.x + threadIdx.x;
  if (p >= HW_) return;

  // Streaming reads: non-temporal (do not pollute L2; atlas needs it).
  const int q = __builtin_nontemporal_load(quad + p);
  const v2f* __restrict__ uv2 = (const v2f*)uv;
  const v2f t = __builtin_nontemporal_load(uv2 + p);
  const float u = t.x;
  const float v = t.y;

  const float u0f = floorf(u);
  const float v0f = floorf(v);
  const float du = u - u0f;
  const float dv = v - v0f;

  int u0 = (int)u0f;
  int v0 = (int)v0f;
  u0 = min(max(u0, 0), N_ - 1);
  v0 = min(max(v0, 0), N_ - 1);
  const int u1 = min(u0 + 1, N_ - 1);
  const int v1 = min(v0 + 1, N_ - 1);

  const float w00 = (1.0f - du) * (1.0f - dv);
  const float w01 = du * (1.0f - dv);
  const float w10 = (1.0f - du) * dv;
  const float w11 = du * dv;

  // Flat texel offsets within one (b,c) plane; max 5,242,879 so 32-bit safe.
  const int base = q * (N_ * N_);
  const int vr0  = base + v0 * N_;
  const int vr1  = base + v1 * N_;
  const int o00 = vr0 + u0;
  const int o01 = vr0 + u1;
  const int o10 = vr1 + u0;
  const int o11 = vr1 + u1;

  // Fully unrolled: 12 planes x 4 taps = 48 loads in flight (LOADcnt <= 63).
  float acc[BC_];
#pragma unroll
  for (int bc = 0; bc < BC_; ++bc) {
    const float* __restrict__ xf = x + (size_t)bc * (size_t)FNN_;
    const float t00 = xf[o00];
    const float t01 = xf[o01];
    const float t10 = xf[o10];
    const float t11 = xf[o11];
    acc[bc] = fmaf(t00, w00, fmaf(t01, w01, fmaf(t10, w10, t11 * w11)));
  }

  // 12 coalesced non-temporal store streams (output is write-once).
#pragma unroll
  for (int bc = 0; bc < BC_; ++bc) {
    __builtin_nontemporal_store(acc[bc], out + (size_t)bc * (size_t)HW_ + p);
  }
}

extern "C" void kernel_launch(void* const* d_in, const int* in_sizes, int n_in,
                              void* d_out, int out_size, void* d_ws, size_t ws_size,
                              hipStream_t stream) {
  (void)in_sizes; (void)n_in; (void)d_ws; (void)ws_size; (void)out_size;
  const float* x    = (const float*)d_in[0];  // [B, C, F, N, N] f32
  const int*   quad = (const int*)  d_in[1];  // [H, W] i32
  const float* uv   = (const float*)d_in[2];  // [H, W, 2] f32
  float*       out  = (float*)d_out;          // [B, C, H, W] f32

  const int threads = 256;                    // 8 wave32 waves per block
  const int blocks  = (HW_ + threads - 1) / threads;  // 8192 blocks
  tangent_to_equirect_bilinear<<<blocks, threads, 0, stream>>>(x, quad, uv, out);
}